// TadInsulationScoreLayer_24154896072775
// MI455X (gfx1250) — compile-verified
//
#include <hip/hip_runtime.h>
#include <hip/hip_bf16.h>

typedef __attribute__((ext_vector_type(2))) float v2f;
typedef __attribute__((ext_vector_type(8))) float v8f;

namespace {
constexpr int kMat          = 512;
constexpr int kD            = 10;
constexpr int kNR           = kMat - 2 * kD;        // 492 outputs per matrix
constexpr int kTilesPerMat  = (kNR + 15) / 16;      // 31 tiles of 16 outputs
constexpr int kBatch        = 256;
constexpr int kWavesPerBlk  = 8;
constexpr int kTotalTiles   = kBatch * kTilesPerMat; // 7936 (divisible by 8)
}

// One wave32 per (batch, t-tile). For tile base t0 and u = t - t0:
//   X[r,c] = x[b, t0 + r, t0 + 11 + c]          (r,c in [0,25), K padded to 28)
//   Q = X * Mask^T  via v_wmma_f32_16x16x4_f32, Mask[c,u] = (u <= c <= u+9)
//   out[t0+u] = (1/100) * sum_{r=u}^{u+9} Q[r,u]  (in-lane predicated adds + xor-16 shuffle)
__global__ __launch_bounds__(kWavesPerBlk * 32)
void diamond_wmma_kernel(const float* __restrict__ x, float* __restrict__ out) {
    const int wave = threadIdx.x >> 5;
    const int lane = threadIdx.x & 31;
    const int tile = blockIdx.x * kWavesPerBlk + wave;
    const int b    = tile / kTilesPerMat;
    const int tt   = tile - b * kTilesPerMat;
    const int t0   = tt * 16;

    const int u = lane & 15;   // A-matrix row M; B/D column N (output index in tile)
    const int h = lane >> 4;   // lane half selects the K sub-pair

    const float* __restrict__ xb   = x + (size_t)b * (kMat * kMat);
    const float* __restrict__ row0 = xb + (size_t)(t0 + u)      * kMat; // X rows 0..15
    const float* __restrict__ row1 = xb + (size_t)(t0 + 16 + u) * kMat; // X rows 16..31 (16..24 live)

    v8f q0 = {}; // Q rows 0..15
    v8f q1 = {}; // Q rows 16..31

    const int colbase = t0 + kD + 1; // t0 + 11

    #pragma unroll
    for (int kk = 0; kk < 7; ++kk) {
        // A 16x4 f32 layout: reg p, lane-half h -> K = 4*kk + p + 2h (contiguous pair per lane)
        const int c0 = kk * 4 + 2 * h;
        const int c1 = c0 + 1;

        // B operand: column-window mask, built in registers (no memory traffic)
        v2f bm;
        bm.x = (c0 >= u && c0 <= u + (kD - 1)) ? 1.0f : 0.0f;
        bm.y = (c1 >= u && c1 <= u + (kD - 1)) ? 1.0f : 0.0f;

        // A operands: clamp address, select-zero out-of-matrix columns (finite data -> 0*x == 0)
        const int g0 = colbase + c0;
        const int g1 = colbase + c1;
        const int s0 = g0 < kMat ? g0 : kMat - 1;
        const int s1 = g1 < kMat ? g1 : kMat - 1;

        const float t00 = row0[s0], t01 = row0[s1];
        const float t10 = row1[s0], t11 = row1[s1];

        v2f a0, a1;
        a0.x = (g0 < kMat) ? t00 : 0.0f;
        a0.y = (g1 < kMat) ? t01 : 0.0f;
        a1.x = (g0 < kMat) ? t10 : 0.0f;
        a1.y = (g1 < kMat) ? t11 : 0.0f;

        q0 = __builtin_amdgcn_wmma_f32_16x16x4_f32(false, a0, false, bm, (short)0, q0, false, false);
        q1 = __builtin_amdgcn_wmma_f32_16x16x4_f32(false, a1, false, bm, (short)0, q1, false, false);
    }

    // C/D layout: lane L holds column N = L&15, rows M = v + 8*(L>>4) in VGPR v.
    // Row window u..u+9 -> predicated in-lane adds, then combine the two lane halves.
    float partial = 0.0f;
    #pragma unroll
    for (int v = 0; v < 8; ++v) {
        const int r0 = v + 8 * h;
        const int r1 = 16 + v + 8 * h;
        partial += (r0 >= u && r0 <= u + (kD - 1)) ? q0[v] : 0.0f;
        partial += (r1 >= u && r1 <= u + (kD - 1)) ? q1[v] : 0.0f;
    }
    const float total = partial + __shfl_xor(partial, 16, 32);

    if (h == 0) {
        const int t = t0 + u;
        if (t < kNR) out[(size_t)b * kNR + t] = total * (1.0f / 100.0f);
    }
}

extern "C" void kernel_launch(void* const* d_in, const int* in_sizes, int n_in,
                              void* d_out, int out_size, void* d_ws, size_t ws_size,
                              hipStream_t stream) {
    (void)in_sizes; (void)n_in; (void)d_ws; (void)ws_size; (void)out_size;
    const float* x = (const float*)d_in[0];
    float* out    = (float*)d_out;
    const int blocks = kTotalTiles / kWavesPerBlk; // 992
    diamond_wmma_kernel<<<blocks, kWavesPerBlk * 32, 0, stream>>>(x, out);
}